// MonotonicAligner_78039555768479
// MI455X (gfx1250) — compile-verified
//
#include <hip/hip_runtime.h>

// MonotonicAligner for MI455X (gfx1250, wave32).
// Phase 1: batched GEMM nlp = -(mel @ text^T) via V_WMMA_F32_16X16X4_F32 (fp32-exact).
// Phase 2: per-batch MAS row-scan DP with decision bits ballot-packed into LDS,
//          then an LDS-resident backtrace (no cost matrix in HBM at all).

typedef __attribute__((ext_vector_type(2))) float v2f;
typedef __attribute__((ext_vector_type(8))) float v8f;

#define B_  32
#define M_  1000   // T_mel
#define N_  200    // T_text
#define K_  512
#define MT_ 63     // ceil(1000/16)
#define NT_ 13     // ceil(200/16)
#define DEC_W 7    // ceil(200/32) dwords of decision bits per row

// ---------------------------------------------------------------------------
// Kernel 1: nlp[b, m, n] = -dot(mel[b,m,:], text[b,n,:])
// One wave per 16x16 output tile; K swept 4 at a time with f32 WMMA.
// ---------------------------------------------------------------------------
__global__ __launch_bounds__(256) void mas_gemm_nlp(
    const float* __restrict__ text,   // (B, N, K)
    const float* __restrict__ mel,    // (B, M, K)
    float* __restrict__ nlp)          // (B, M, N)
{
    const int wid  = threadIdx.x >> 5;
    const int lane = threadIdx.x & 31;
    const int t    = blockIdx.x * 8 + wid;           // wave-uniform tile id
    if (t >= B_ * MT_ * NT_) return;                 // uniform per wave: EXEC stays all-1s

    const int b   = t / (MT_ * NT_);
    const int rem = t % (MT_ * NT_);
    const int m0  = (rem / NT_) * 16;
    const int n0  = (rem % NT_) * 16;

    const int lrow  = lane & 15;
    const int khalf = (lane >> 4) * 2;               // lanes 16-31 carry K+2,K+3

    // Clamp edge-tile rows for loads (stores are masked, so clamped data is discarded).
    int mrow = m0 + lrow; if (mrow > M_ - 1) mrow = M_ - 1;
    int nrow = n0 + lrow; if (nrow > N_ - 1) nrow = N_ - 1;

    const float* aptr = mel  + ((size_t)b * M_ + mrow) * K_ + khalf;
    const float* bptr = text + ((size_t)b * N_ + nrow) * K_ + khalf;

    v8f c = {};
#pragma unroll 4
    for (int k0 = 0; k0 < K_; k0 += 4) {
        v2f a  = *(const v2f*)(aptr + k0);
        v2f bb = *(const v2f*)(bptr + k0);
        // D = A(16x4 f32) * B(4x16 f32) + C(16x16 f32)
        c = __builtin_amdgcn_wmma_f32_16x16x4_f32(
                /*neg_a=*/false, a, /*neg_b=*/false, bb,
                /*c_mod=*/(short)0, c, /*reuse_a=*/false, /*reuse_b=*/false);
    }

    // C/D layout: VGPR r, lane l -> row m0 + r + (l>=16 ? 8 : 0), col n0 + (l&15)
    float* out = nlp + (size_t)b * M_ * N_;
    const int colN  = n0 + lrow;
    const int rbase = m0 + ((lane >> 4) * 8);
    if (colN < N_) {
#pragma unroll
        for (int r = 0; r < 8; ++r) {
            const int row = rbase + r;
            if (row < M_) out[(size_t)row * N_ + colN] = -c[r];
        }
    }
}

// ---------------------------------------------------------------------------
// Kernel 2: per-batch MAS DP + backtrace. One 256-thread workgroup per batch.
// prev/cur rows double-buffered in LDS (one barrier per row); the diag<=up
// decision bit per cell is ballot-packed (wave32) into 7 dwords per row
// (28 KB total, well inside 320 KB/WGP). Backtrace then runs from LDS.
// ---------------------------------------------------------------------------
__global__ __launch_bounds__(256) void mas_dp_backtrace(
    const float* __restrict__ nlp,    // (B, M, N)
    float* __restrict__ path)         // (B, M, N), pre-zeroed
{
    __shared__ float buf0[N_];
    __shared__ float buf1[N_];
    __shared__ unsigned int dec[M_ * DEC_W];   // dec[i][j]: cost[i-1,j-1] <= cost[i-1,j]

    const int b   = blockIdx.x;
    const int tid = threadIdx.x;
    const int wid  = tid >> 5;
    const int lane = tid & 31;
    const float INF = __builtin_huge_valf();

    const float* np = nlp  + (size_t)b * M_ * N_;
    float*       pp = path + (size_t)b * M_ * N_;

    float* prev = buf0;
    float* cur  = buf1;

    // cost row 0 = [nlp[0,0], inf, inf, ...]
    if (tid < N_) prev[tid] = (tid == 0) ? np[0] : INF;

    // software prefetch of the next nlp row (hides L2 latency across the barrier)
    float nv = (tid < N_) ? np[N_ + tid] : 0.0f;
    __syncthreads();

    for (int i = 1; i < M_; ++i) {
        const float v = nv;
        if (i + 1 < M_ && tid < N_) nv = np[(size_t)(i + 1) * N_ + tid];  // issue early

        const float pl  = (tid > 0 && tid < N_) ? prev[tid - 1] : INF;
        const float p   = (tid < N_)            ? prev[tid]     : INF;
        const bool diag = (pl <= p);
        const unsigned long long bal = __ballot(diag);

        if (tid < N_) cur[tid] = v + fminf(pl, p);
        if (wid < DEC_W && lane == 0) dec[i * DEC_W + wid] = (unsigned int)bal;
        __syncthreads();

        float* tmp = prev; prev = cur; cur = tmp;
    }

    // Backtrace (lane 0 of the workgroup), entirely from LDS decision bits.
    if (tid == 0) {
        int i = M_ - 1, j = N_ - 1;
        pp[(size_t)i * N_ + j] = 1.0f;
        for (int s = 0; s < M_ + N_ - 2; ++s) {
            const bool at_origin = (i == 0) && (j == 0);
            bool diag_better = false;
            if (i > 0 && j > 0)
                diag_better = (dec[i * DEC_W + (j >> 5)] >> (j & 31)) & 1u;
            const int di = (at_origin || i == 0) ? 0 : 1;
            const int dj = (at_origin || j == 0) ? 0 : ((i == 0 || diag_better) ? 1 : 0);
            i -= di; j -= dj;
            pp[(size_t)i * N_ + j] = 1.0f;   // idempotent re-writes at origin are fine
        }
    }
}

// ---------------------------------------------------------------------------
extern "C" void kernel_launch(void* const* d_in, const int* in_sizes, int n_in,
                              void* d_out, int out_size, void* d_ws, size_t ws_size,
                              hipStream_t stream) {
    const float* text = (const float*)d_in[0];   // (32, 200, 512) f32
    const float* mel  = (const float*)d_in[1];   // (32, 1000, 512) f32
    float* out = (float*)d_out;                  // (32, 1000, 200) f32
    float* nlp = (float*)d_ws;                   // needs 32*1000*200*4 = 25.6 MB

    // Zero the path output (capturable memset node; guarantees 0-fill completes
    // before the backtrace's 1.0 stores since kernels serialize on the stream).
    hipMemsetAsync(d_out, 0, (size_t)out_size * sizeof(float), stream);

    // Phase 1: 32*63*13 = 26208 wave-tiles, 8 waves per 256-thread block.
    const int tiles  = B_ * MT_ * NT_;
    const int blocks = (tiles + 7) / 8;          // = 3276 exactly
    mas_gemm_nlp<<<blocks, 256, 0, stream>>>(text, mel, nlp);

    // Phase 2: one workgroup per batch.
    mas_dp_backtrace<<<B_, 256, 0, stream>>>(nlp, out);
}